// HybridCuriosityModule_77017353552468
// MI455X (gfx1250) — compile-verified
//
#include <hip/hip_runtime.h>
#include <hip/hip_bf16.h>
#include <float.h>

typedef __attribute__((ext_vector_type(2))) float v2f;
typedef __attribute__((ext_vector_type(4))) float v4f;
typedef __attribute__((ext_vector_type(8))) float v8f;

#define EPSF 1e-8f

__device__ __forceinline__ float wave_sum(float v) {
#pragma unroll
  for (int m = 16; m >= 1; m >>= 1) v += __shfl_xor(v, m, 32);
  return v;
}

// sum across the 16 lanes of each half-wave (masks < 16 stay inside the half)
__device__ __forceinline__ float half_sum16(float v) {
#pragma unroll
  for (int m = 8; m >= 1; m >>= 1) v += __shfl_xor(v, m, 32);
  return v;
}

// One 16x16 fp32 output tile, D = A(16xK, LDS) * B(KxN, global) cols [n0,n0+16),
// accumulated with V_WMMA_F32_16X16X4_F32 (fp32-exact, K step 4).
// A frag (16x4 f32, 2 VGPR): lanes 0-15 -> A[l][k+0..1], lanes 16-31 -> A[l][k+2..3]
// B frag (4x16 f32, 2 VGPR): v0: lanes0-15 B[k+0][n], lanes16-31 B[k+2][n]
//                            v1: lanes0-15 B[k+1][n], lanes16-31 B[k+3][n]
__device__ __forceinline__ v8f wmma_tile_f32(const float* __restrict__ ldsA, int ldA,
                                             const float* __restrict__ Bg, int ldb,
                                             int n0, int K) {
  const int lane = threadIdx.x & 31;
  const int hi   = lane >> 4;
  const int l16  = lane & 15;
  v8f acc = {};
  for (int k = 0; k < K; k += 4) {
    const float* ap = ldsA + l16 * ldA + (k + 2 * hi);
    v2f a; a.x = ap[0]; a.y = ap[1];                       // ds_load_b64
    const float* bp = Bg + (size_t)(k + 2 * hi) * ldb + n0 + l16;
    v2f b; b.x = bp[0]; b.y = bp[ldb];
    acc = __builtin_amdgcn_wmma_f32_16x16x4_f32(false, a, false, b,
                                                (short)0, acc, false, false);
  }
  return acc;
}

// C/D layout (32-bit 16x16): VGPR r, lanes 0-15 -> row r, lanes 16-31 -> row r+8
__device__ __forceinline__ void store_tile(float* __restrict__ out, int ldo, int n0,
                                           v8f acc, const float* __restrict__ bias,
                                           bool do_relu) {
  const int lane = threadIdx.x & 31;
  const int hi   = lane >> 4;
  const int l16  = lane & 15;
  const float bv = bias[n0 + l16];
#pragma unroll
  for (int r = 0; r < 8; ++r) {
    float v = acc[r] + bv;
    if (do_relu) v = fmaxf(v, 0.0f);
    out[(r + 8 * hi) * ldo + n0 + l16] = v;
  }
}

// ---------------- RND bonus: fused target + predictor MLPs ----------------
// grid = 16 blocks (16 batch rows each), block = 256 threads = 8 waves.
__global__ __launch_bounds__(256) void rnd_kernel(
    const float* __restrict__ state,
    const float* __restrict__ tW1, const float* __restrict__ tb1,
    const float* __restrict__ tW2, const float* __restrict__ tb2,
    const float* __restrict__ pW1, const float* __restrict__ pb1,
    const float* __restrict__ pW2, const float* __restrict__ pb2,
    const float* __restrict__ pW3, const float* __restrict__ pb3,
    float* __restrict__ rnd_out) {
  __shared__ float sX[16 * 512];   // X tile, later reused as h2 buffer
  __shared__ float sH[16 * 256];   // hidden activations
  __shared__ float sT[16 * 128];   // target features
  __shared__ float sPart[8][16];   // per-wave, per-row partial SSE

  const int tid  = threadIdx.x;
  const int wave = tid >> 5;
  const int lane = tid & 31;
  const int hi   = lane >> 4;
  const int l16  = lane & 15;
  const int row0 = blockIdx.x * 16;

  // stage 16 contiguous rows of X (16x512 f32 = 32KB), fully coalesced
  for (int i = tid; i < 16 * 512; i += 256)
    sX[i] = state[(size_t)row0 * 512 + i];
  __syncthreads();

  // target layer 1: relu(X @ tW1 + tb1) -> sH (16x256)
  for (int t = wave; t < 16; t += 8)
    store_tile(sH, 256, t * 16, wmma_tile_f32(sX, 512, tW1, 256, t * 16, 512), tb1, true);
  __syncthreads();

  // target layer 2: sH @ tW2 + tb2 -> sT (16x128), one tile per wave
  store_tile(sT, 128, wave * 16, wmma_tile_f32(sH, 256, tW2, 128, wave * 16, 256), tb2, false);
  __syncthreads();

  // predictor layer 1: relu(X @ pW1 + pb1) -> sH (overwrite, target h consumed)
  for (int t = wave; t < 16; t += 8)
    store_tile(sH, 256, t * 16, wmma_tile_f32(sX, 512, pW1, 256, t * 16, 512), pb1, true);
  __syncthreads();

  // predictor layer 2: relu(sH @ pW2 + pb2) -> reuse sX as h2 (16x256)
  float* sH2 = sX;
  for (int t = wave; t < 16; t += 8)
    store_tile(sH2, 256, t * 16, wmma_tile_f32(sH, 256, pW2, 256, t * 16, 256), pb2, true);
  __syncthreads();

  // predictor layer 3 + squared error vs target, deterministic butterfly reduce
  {
    const int n0 = wave * 16;
    v8f acc = wmma_tile_f32(sH2, 256, pW3, 128, n0, 256);
    const float bv = pb3[n0 + l16];
#pragma unroll
    for (int r = 0; r < 8; ++r) {
      const int row = r + 8 * hi;
      float d  = (acc[r] + bv) - sT[row * 128 + n0 + l16];
      float tot = half_sum16(d * d);          // sum over this tile's 16 cols
      if (l16 == 0) sPart[wave][row] = tot;
    }
  }
  __syncthreads();

  if (tid < 16) {
    float s = 0.0f;
#pragma unroll
    for (int w = 0; w < 8; ++w) s += sPart[w][tid];   // fixed order: deterministic
    rnd_out[row0 + tid] = s * (1.0f / 128.0f);
  }
}

// ---------------- NGU: single-pass NT stream over memory[M,256,512] ----------------
// grid = 256 blocks (one batch column b), 8 waves stride over m; one wave per row.
__global__ __launch_bounds__(256) void ngu_kernel(
    const float* __restrict__ state, const float* __restrict__ memory,
    int M, float* __restrict__ pmin) {
  __shared__ float sS[512];
  __shared__ float sRed[8];
  __shared__ float sMin[8];
  __shared__ float sInv;

  const int tid  = threadIdx.x;
  const int wave = tid >> 5;
  const int lane = tid & 31;
  const int b    = blockIdx.x;

  sS[tid]       = state[(size_t)b * 512 + tid];
  sS[tid + 256] = state[(size_t)b * 512 + tid + 256];
  __syncthreads();

  // 1 / max(||state_b||, eps)
  float loc = sS[tid] * sS[tid] + sS[tid + 256] * sS[tid + 256];
  loc = wave_sum(loc);
  if (lane == 0) sRed[wave] = loc;
  __syncthreads();
  if (tid == 0) {
    float ssq = 0.0f;
    for (int w = 0; w < 8; ++w) ssq += sRed[w];
    sInv = 1.0f / fmaxf(sqrtf(ssq), EPSF);
  }
  __syncthreads();

  const v4f* s4 = reinterpret_cast<const v4f*>(sS);
  float wmin = FLT_MAX;
  for (int m = wave; m < M; m += 8) {
    const v4f* m4 = reinterpret_cast<const v4f*>(memory + ((size_t)m * 256 + b) * 512);
    float dot = 0.0f, msq = 0.0f;
#pragma unroll
    for (int j = 0; j < 4; ++j) {                 // 4 x global_load_b128 NT per lane
      v4f mv = __builtin_nontemporal_load(m4 + lane + j * 32);
      v4f sv = s4[lane + j * 32];
      dot += mv.x * sv.x + mv.y * sv.y + mv.z * sv.z + mv.w * sv.w;
      msq += mv.x * mv.x + mv.y * mv.y + mv.z * mv.z + mv.w * mv.w;
    }
    dot = wave_sum(dot);
    msq = wave_sum(msq);
    if (lane == 0) {
      float sim = dot * sInv / fmaxf(sqrtf(msq), EPSF);
      wmin = fminf(wmin, sim);
    }
  }
  if (lane == 0) sMin[wave] = wmin;
  __syncthreads();
  if (tid == 0) {
    float mn = FLT_MAX;
    for (int w = 0; w < 8; ++w) mn = fminf(mn, sMin[w]);
    pmin[b] = mn;
  }
}

// ---------------- finalize: global min -> ngu scalar, combine ----------------
__global__ __launch_bounds__(256) void finalize_kernel(
    const float* __restrict__ rnd, const float* __restrict__ pmin,
    float* __restrict__ out) {
  __shared__ float sMin[8];
  __shared__ float sNgu;
  const int tid  = threadIdx.x;
  const int wave = tid >> 5;
  const int lane = tid & 31;

  float v = pmin[tid];
#pragma unroll
  for (int m = 16; m >= 1; m >>= 1) v = fminf(v, __shfl_xor(v, m, 32));
  if (lane == 0) sMin[wave] = v;
  __syncthreads();
  if (tid == 0) {
    float mn = FLT_MAX;
    for (int w = 0; w < 8; ++w) mn = fminf(mn, sMin[w]);
    sNgu = 1.0f / (1.0f + mn);
  }
  __syncthreads();
  out[tid] = 0.5f * rnd[tid] + 0.5f * sNgu;
}

extern "C" void kernel_launch(void* const* d_in, const int* in_sizes, int n_in,
                              void* d_out, int out_size, void* d_ws, size_t ws_size,
                              hipStream_t stream) {
  (void)n_in; (void)out_size; (void)ws_size;
  const float* state = (const float*)d_in[0];
  const float* mem   = (const float*)d_in[1];
  const float* tW1   = (const float*)d_in[2];
  const float* tb1   = (const float*)d_in[3];
  const float* tW2   = (const float*)d_in[4];
  const float* tb2   = (const float*)d_in[5];
  const float* pW1   = (const float*)d_in[6];
  const float* pb1   = (const float*)d_in[7];
  const float* pW2   = (const float*)d_in[8];
  const float* pb2   = (const float*)d_in[9];
  const float* pW3   = (const float*)d_in[10];
  const float* pb3   = (const float*)d_in[11];
  float* out  = (float*)d_out;
  float* ws   = (float*)d_ws;
  float* rnd  = ws;         // 256 floats
  float* pmin = ws + 256;   // 256 floats

  const int M = in_sizes[1] / (256 * 512);   // 1000

  rnd_kernel<<<16, 256, 0, stream>>>(state, tW1, tb1, tW2, tb2,
                                     pW1, pb1, pW2, pb2, pW3, pb3, rnd);
  ngu_kernel<<<256, 256, 0, stream>>>(state, mem, M, pmin);
  finalize_kernel<<<1, 256, 0, stream>>>(rnd, pmin, out);
}